// QuantizedEmbeddingBackbone_33870112096418
// MI455X (gfx1250) — compile-verified
//
#include <hip/hip_runtime.h>

// CDNA5 (gfx1250) wave32 WMMA vector-quantization kernel.
// Score s = p.k - 0.5*||k||^2 via V_WMMA_F32_16X16X4_F32 with K=4 trick:
//   A row  = [px, py, pz, 1.0]
//   B col  = [kx, ky, kz, -0.5*||k||^2]
// argmax_j s  ==  argmin_j ||p - k_j||^2.

typedef __attribute__((ext_vector_type(2))) float v2f;
typedef __attribute__((ext_vector_type(8))) float v8f;

#define M_KEYS 4096
#define D_FEAT 64
#define BN     (4 * 16384)          // B*N points
#define WAVES_PER_BLOCK 8           // 256 threads, wave32
#define POINTS_PER_WAVE 16

__global__ __launch_bounds__(256)
void qvq_wmma_kernel(const float* __restrict__ pc,    // (BN, 3)
                     const float* __restrict__ keys,  // (M_KEYS, 3)
                     const float* __restrict__ vals,  // (M_KEYS, 64)
                     float* __restrict__ out)         // feats | ids | pc
{
    // Preprocessed key table: (kx, ky, kz, -0.5*ksq). 64 KB LDS (WGP has 320 KB).
    __shared__ float4 sk[M_KEYS];

    const int tid = threadIdx.x;
    for (int j = tid; j < M_KEYS; j += 256) {
        float kx = keys[j * 3 + 0];
        float ky = keys[j * 3 + 1];
        float kz = keys[j * 3 + 2];
        float ksq = kx * kx + ky * ky + kz * kz;
        sk[j] = make_float4(kx, ky, kz, -0.5f * ksq);
    }
    __syncthreads();

    const int wave = tid >> 5;
    const int lane = tid & 31;
    const int col  = lane & 15;          // N position (key column / point row id)
    const bool hi  = lane >= 16;         // upper lane half -> K=2,3 of A/B
    const int tile = blockIdx.x * WAVES_PER_BLOCK + wave;
    const int base = tile * POINTS_PER_WAVE;   // first point of this wave's tile

    // A-matrix (16x4 f32, ISA layout): lanes 0-15 hold (K0,K1)=(px,py) of row M=lane,
    // lanes 16-31 hold (K2,K3)=(pz, 1.0) of row M=lane-16.
    v2f a;
    {
        const float* p = pc + (size_t)(base + col) * 3;
        if (hi) { a.x = p[2]; a.y = 1.0f; }
        else    { a.x = p[0]; a.y = p[1]; }
    }

    // Running argmax over keys. C/D layout: VGPR i = row M=i (lanes 0-15) or
    // M=i+8 (lanes 16-31), column N = lane&15.
    float bestv[8];
    int   besti[8];
#pragma unroll
    for (int i = 0; i < 8; ++i) { bestv[i] = -3.402823466e38f; besti[i] = 0; }

#pragma unroll 4
    for (int t = 0; t < M_KEYS / 16; ++t) {
        const int j = t * 16 + col;                 // this lane's key column
        float4 k4 = sk[j];                          // ds_load_b128, halves broadcast
        v2f b;
        if (hi) { b.x = k4.z; b.y = k4.w; }         // (kz, -0.5*ksq) -> K2,K3
        else    { b.x = k4.x; b.y = k4.y; }         // (kx, ky)      -> K0,K1

        v8f c = {};
        // 8 args: (neg_a, A, neg_b, B, c_mod, C, reuse_a, reuse_b)
        c = __builtin_amdgcn_wmma_f32_16x16x4_f32(
                false, a, false, b, (short)0, c, false, false);

#pragma unroll
        for (int i = 0; i < 8; ++i) {
            // strict '>' keeps the first (lowest-index) maximum -> matches argmin
            if (c[i] > bestv[i]) { bestv[i] = c[i]; besti[i] = j; }
        }
    }

    // Reduce (val, idx) across the 16 lanes of each half (masks < 16 stay in-half).
#pragma unroll
    for (int off = 8; off >= 1; off >>= 1) {
#pragma unroll
        for (int i = 0; i < 8; ++i) {
            float ov = __shfl_xor(bestv[i], off, 32);
            int   oi = __shfl_xor(besti[i], off, 32);
            if (ov > bestv[i] || (ov == bestv[i] && oi < besti[i])) {
                bestv[i] = ov; besti[i] = oi;
            }
        }
    }

    float* feats = out;                               // BN * 64
    float* idout = out + (size_t)BN * D_FEAT;         // BN
    float* pcout = idout + BN;                        // BN * 3

    // Feature gather: row r maps to point base + r (+8 for upper half).
    // 16 lanes x float4 cover one 64-float embedding row, fully coalesced.
#pragma unroll
    for (int r = 0; r < 8; ++r) {
        const int gp = base + r + (hi ? 8 : 0);
        const float4* src = (const float4*)(vals + (size_t)besti[r] * D_FEAT);
        float4* dst = (float4*)(feats + (size_t)gp * D_FEAT);
        dst[col] = src[col];
    }

    // ids (stored as float: numpy concat of int ids with float arrays upcasts).
    if (col == 0) {
#pragma unroll
        for (int r = 0; r < 8; ++r)
            idout[base + r + (hi ? 8 : 0)] = (float)besti[r];
    }

    // Pointcloud passthrough: lanes 0-15 each copy one point's 3 floats.
    if (!hi) {
        const size_t gp = (size_t)(base + col);
        pcout[gp * 3 + 0] = pc[gp * 3 + 0];
        pcout[gp * 3 + 1] = pc[gp * 3 + 1];
        pcout[gp * 3 + 2] = pc[gp * 3 + 2];
    }
}

extern "C" void kernel_launch(void* const* d_in, const int* in_sizes, int n_in,
                              void* d_out, int out_size, void* d_ws, size_t ws_size,
                              hipStream_t stream) {
    const float* pc   = (const float*)d_in[0];   // (B,N,3)
    const float* keys = (const float*)d_in[1];   // (M,3)
    const float* vals = (const float*)d_in[2];   // (M,64)
    float* out = (float*)d_out;

    const int points = BN;                                        // 65536
    const int blocks = points / (POINTS_PER_WAVE * WAVES_PER_BLOCK); // 512
    qvq_wmma_kernel<<<blocks, 256, 0, stream>>>(pc, keys, vals, out);
}